// LinkPredictionModel_79963701117029
// MI455X (gfx1250) — compile-verified
//
#include <hip/hip_runtime.h>
#include <hip/hip_bf16.h>
#include <math.h>

typedef __attribute__((ext_vector_type(2))) float v2f;
typedef __attribute__((ext_vector_type(8))) float v8f;

#define N_NODES   100000
#define NUM_FEAT  128
#define H1_DIM    16
#define H2_DIM    32

// ---------------------------------------------------------------------------
// Degree / normalization
// ---------------------------------------------------------------------------
__global__ void deg_init_kernel(float* __restrict__ deg, int n) {
    int i = blockIdx.x * blockDim.x + threadIdx.x;
    if (i < n) deg[i] = 1.0f;   // self-loop contributes 1 to every node's degree
}

__global__ void deg_edges_kernel(const int* __restrict__ dst, float* __restrict__ deg, int E) {
    int e = blockIdx.x * blockDim.x + threadIdx.x;
    if (e < E) atomicAdd(&deg[dst[e]], 1.0f);
}

__global__ void dinv_kernel(const float* __restrict__ deg, float* __restrict__ dinv, int n) {
    int i = blockIdx.x * blockDim.x + threadIdx.x;
    if (i < n) dinv[i] = rsqrtf(deg[i]);   // deg >= 1 always (self-loops)
}

// ---------------------------------------------------------------------------
// GEMM1: H1[100000,16] = X[100000,128] @ W1[128,16]    (fp32 WMMA 16x16x4)
// One wave computes one 16-row tile; K advanced 4 per WMMA (32 WMMAs).
// A layout: lanes 0-15 -> K={k0,k0+1}, lanes 16-31 -> K={k0+2,k0+3}
// B layout: lane%16 = column n; same K split across lane halves.
// C/D: VGPR r, lanes 0-15 -> row r; lanes 16-31 -> row r+8.
// ---------------------------------------------------------------------------
__global__ void gemm1_kernel(const float* __restrict__ X, const float* __restrict__ W1,
                             float* __restrict__ H1, int nTiles) {
    int wave = (blockIdx.x * blockDim.x + threadIdx.x) >> 5;
    int lane = threadIdx.x & 31;
    if (wave >= nTiles) return;               // wave-uniform: EXEC all-1s for WMMA
    int n16 = lane & 15;                      // A: row within tile / B: column
    int hi  = lane >> 4;                      // K-half selector
    const float* xrow = X + (size_t)(wave * 16 + n16) * NUM_FEAT;

    v8f acc = {0.f, 0.f, 0.f, 0.f, 0.f, 0.f, 0.f, 0.f};
#pragma unroll
    for (int k0 = 0; k0 < NUM_FEAT; k0 += 4) {
        int k = k0 + 2 * hi;
        v2f a; a.x = xrow[k];                 a.y = xrow[k + 1];
        v2f b; b.x = W1[k * H1_DIM + n16];    b.y = W1[(k + 1) * H1_DIM + n16];
        acc = __builtin_amdgcn_wmma_f32_16x16x4_f32(false, a, false, b,
                                                    (short)0, acc, false, false);
    }
    float* out = H1 + (size_t)(wave * 16) * H1_DIM;
#pragma unroll
    for (int r = 0; r < 8; ++r)
        out[(r + 8 * hi) * H1_DIM + n16] = acc[r];
}

// ---------------------------------------------------------------------------
// Self-loop init: agg[i,f] = dinv[i]^2 * h[i,f]   (elementwise over N*F)
// ---------------------------------------------------------------------------
__global__ void selfloop_init_kernel(const float* __restrict__ h, const float* __restrict__ dinv,
                                     float* __restrict__ agg, int total, int logF) {
    int idx = blockIdx.x * blockDim.x + threadIdx.x;
    if (idx >= total) return;
    int node = idx >> logF;
    float d = dinv[node];
    agg[idx] = d * d * h[idx];
}

// ---------------------------------------------------------------------------
// Edge scatter: agg[dst] += dinv[src]*dinv[dst] * h[src]
// 4 threads per edge (16 feat) or 8 threads per edge (32 feat); float4 chunks.
// ---------------------------------------------------------------------------
__global__ void scatter16_kernel(const int* __restrict__ src, const int* __restrict__ dst,
                                 const float* __restrict__ dinv, const float* __restrict__ h,
                                 float* __restrict__ agg, int E) {
    long long t = (long long)blockIdx.x * blockDim.x + threadIdx.x;
    int e = (int)(t >> 2);
    if (e >= E) return;
    int f = ((int)t & 3) * 4;
    int s = src[e], d = dst[e];
    float norm = dinv[s] * dinv[d];
    const float4 v = *(const float4*)(h + (size_t)s * H1_DIM + f);
    float* o = agg + (size_t)d * H1_DIM + f;
    atomicAdd(o + 0, norm * v.x);
    atomicAdd(o + 1, norm * v.y);
    atomicAdd(o + 2, norm * v.z);
    atomicAdd(o + 3, norm * v.w);
}

__global__ void scatter32_kernel(const int* __restrict__ src, const int* __restrict__ dst,
                                 const float* __restrict__ dinv, const float* __restrict__ h,
                                 float* __restrict__ agg, int E) {
    long long t = (long long)blockIdx.x * blockDim.x + threadIdx.x;
    int e = (int)(t >> 3);
    if (e >= E) return;
    int f = ((int)t & 7) * 4;
    int s = src[e], d = dst[e];
    float norm = dinv[s] * dinv[d];
    const float4 v = *(const float4*)(h + (size_t)s * H2_DIM + f);
    float* o = agg + (size_t)d * H2_DIM + f;
    atomicAdd(o + 0, norm * v.x);
    atomicAdd(o + 1, norm * v.y);
    atomicAdd(o + 2, norm * v.z);
    atomicAdd(o + 3, norm * v.w);
}

// ---------------------------------------------------------------------------
// GEMM2: H2[100000,32] = relu(agg1+b1)[100000,16] @ W2[16,32]
// One wave per 16-row tile; two accumulators for the two 16-col halves.
// K=16 -> 4 WMMA steps * 2 accumulators = 8 WMMAs per wave.
// ---------------------------------------------------------------------------
__global__ void gemm2_kernel(const float* __restrict__ agg1, const float* __restrict__ b1,
                             const float* __restrict__ W2, float* __restrict__ H2, int nTiles) {
    int wave = (blockIdx.x * blockDim.x + threadIdx.x) >> 5;
    int lane = threadIdx.x & 31;
    if (wave >= nTiles) return;
    int n16 = lane & 15;
    int hi  = lane >> 4;
    const float* arow = agg1 + (size_t)(wave * 16 + n16) * H1_DIM;

    v8f acc0 = {0.f, 0.f, 0.f, 0.f, 0.f, 0.f, 0.f, 0.f};
    v8f acc1 = {0.f, 0.f, 0.f, 0.f, 0.f, 0.f, 0.f, 0.f};
#pragma unroll
    for (int k0 = 0; k0 < H1_DIM; k0 += 4) {
        int k = k0 + 2 * hi;
        v2f a;
        a.x = fmaxf(arow[k]     + b1[k],     0.0f);   // fused bias + ReLU
        a.y = fmaxf(arow[k + 1] + b1[k + 1], 0.0f);
        v2f bA, bB;
        bA.x = W2[k * H2_DIM + n16];          bA.y = W2[(k + 1) * H2_DIM + n16];
        bB.x = W2[k * H2_DIM + 16 + n16];     bB.y = W2[(k + 1) * H2_DIM + 16 + n16];
        acc0 = __builtin_amdgcn_wmma_f32_16x16x4_f32(false, a, false, bA,
                                                     (short)0, acc0, false, false);
        acc1 = __builtin_amdgcn_wmma_f32_16x16x4_f32(false, a, false, bB,
                                                     (short)0, acc1, false, false);
    }
    float* out = H2 + (size_t)(wave * 16) * H2_DIM;
#pragma unroll
    for (int r = 0; r < 8; ++r) {
        out[(r + 8 * hi) * H2_DIM + n16]      = acc0[r];
        out[(r + 8 * hi) * H2_DIM + 16 + n16] = acc1[r];
    }
}

// ---------------------------------------------------------------------------
// Decode: out[m] = sigmoid( (agg2[i]+b2)·Wfc[0:32] + (agg2[j]+b2)·Wfc[32:64] + bfc )
// ---------------------------------------------------------------------------
__global__ void decode_kernel(const float* __restrict__ agg2, const float* __restrict__ b2,
                              const float* __restrict__ Wfc, const float* __restrict__ bfc,
                              const int* __restrict__ examples, float* __restrict__ out, int M) {
    int m = blockIdx.x * blockDim.x + threadIdx.x;
    if (m >= M) return;
    int i = examples[2 * m];
    int j = examples[2 * m + 1];
    const float* hi_ = agg2 + (size_t)i * H2_DIM;
    const float* hj_ = agg2 + (size_t)j * H2_DIM;
    float acc = bfc[0];
#pragma unroll
    for (int k = 0; k < H2_DIM; ++k) acc += (hi_[k] + b2[k]) * Wfc[k];
#pragma unroll
    for (int k = 0; k < H2_DIM; ++k) acc += (hj_[k] + b2[k]) * Wfc[k + H2_DIM];
    out[m] = 1.0f / (1.0f + expf(-acc));
}

// ---------------------------------------------------------------------------
// Host launcher
// ---------------------------------------------------------------------------
extern "C" void kernel_launch(void* const* d_in, const int* in_sizes, int n_in,
                              void* d_out, int out_size, void* d_ws, size_t ws_size,
                              hipStream_t stream) {
    const float* X        = (const float*)d_in[0];
    const int*   edge     = (const int*)  d_in[1];
    const int*   examples = (const int*)  d_in[2];
    const float* W1       = (const float*)d_in[3];
    const float* b1       = (const float*)d_in[4];
    const float* W2       = (const float*)d_in[5];
    const float* b2       = (const float*)d_in[6];
    const float* Wfc      = (const float*)d_in[7];
    const float* bfc      = (const float*)d_in[8];
    float*       out      = (float*)d_out;

    const int E = in_sizes[1] / 2;            // edge_index is [2, E] row-major
    const int M = in_sizes[2] / 2;            // examples   is [M, 2] row-major
    const int N = N_NODES;
    const int* src = edge;                    // edge_index[0]
    const int* dst = edge + E;                // edge_index[1]

    // Workspace carve-up (256B aligned)
    char*  ws = (char*)d_ws;
    size_t off = 0;
    auto carve = [&](size_t bytes) -> char* {
        char* p = ws + off;
        off = (off + bytes + 255) & ~(size_t)255;
        return p;
    };
    float* deg  = (float*)carve((size_t)N * 4);
    float* dinv = (float*)carve((size_t)N * 4);
    float* H1   = (float*)carve((size_t)N * H1_DIM * 4);
    float* agg1 = (float*)carve((size_t)N * H1_DIM * 4);
    float* H2   = (float*)carve((size_t)N * H2_DIM * 4);
    float* agg2 = (float*)carve((size_t)N * H2_DIM * 4);
    (void)ws_size; (void)n_in; (void)out_size;

    const int BT = 256;                       // 8 waves / block (wave32)
    const int nTiles = N / 16;                // 6250, exact

    // 1) degree + normalization
    deg_init_kernel<<<(N + BT - 1) / BT, BT, 0, stream>>>(deg, N);
    deg_edges_kernel<<<(E + BT - 1) / BT, BT, 0, stream>>>(dst, deg, E);
    dinv_kernel<<<(N + BT - 1) / BT, BT, 0, stream>>>(deg, dinv, N);

    // 2) H1 = X @ W1  (WMMA fp32)
    gemm1_kernel<<<(nTiles * 32 + BT - 1) / BT, BT, 0, stream>>>(X, W1, H1, nTiles);

    // 3) layer-1 aggregation: self-loop init + edge scatter
    {
        int total = N * H1_DIM;
        selfloop_init_kernel<<<(total + BT - 1) / BT, BT, 0, stream>>>(H1, dinv, agg1, total, 4);
        long long t = (long long)E * 4;
        scatter16_kernel<<<(unsigned)((t + BT - 1) / BT), BT, 0, stream>>>(src, dst, dinv, H1, agg1, E);
    }

    // 4) H2 = relu(agg1 + b1) @ W2  (WMMA fp32, fused bias/ReLU)
    gemm2_kernel<<<(nTiles * 32 + BT - 1) / BT, BT, 0, stream>>>(agg1, b1, W2, H2, nTiles);

    // 5) layer-2 aggregation
    {
        int total = N * H2_DIM;
        selfloop_init_kernel<<<(total + BT - 1) / BT, BT, 0, stream>>>(H2, dinv, agg2, total, 5);
        long long t = (long long)E * 8;
        scatter32_kernel<<<(unsigned)((t + BT - 1) / BT), BT, 0, stream>>>(src, dst, dinv, H2, agg2, E);
    }

    // 6) decode (b2 fused here)
    decode_kernel<<<(M + BT - 1) / BT, BT, 0, stream>>>(agg2, b2, Wfc, bfc, examples, out, M);
}